// GODE_90778428768463
// MI455X (gfx1250) — compile-verified
//
#include <hip/hip_runtime.h>
#include <hip/hip_bf16.h>

// GAT layer for MI455X (gfx1250, wave32).
//  N=100000 nodes, E=1.6M edges (+N self loops), IN_F=128, HEADS=4, OUT_F=32.
//
// h = x@W via V_WMMA_F32_16X16X4_F32 (fp32-exact matrix path):
//   - W pre-packed once into B-fragment order (coalesced b64 loads),
//   - each wave owns one 16-row M-tile and all 8 N-tiles => A reused 8x,
//     8 independent accumulator chains hide WMMA RAW hazards.
// Edge softmax + scatter via native f32 global atomics (h/out ~51MB each, L2-resident on 192MB L2).

typedef __attribute__((ext_vector_type(2))) float v2f;
typedef __attribute__((ext_vector_type(8))) float v8f;

#define IN_F   128
#define HxF    128   // HEADS*OUT_F
#define HEADS  4
#define NEG_SLOPE 0.2f

// ---------------------------------------------------------------------------
// Kernel 0: pack W[128,128] into B-fragment layout:
//   Wf[(kk*8 + nT)*32 + lane] = { W[krow][c], W[krow+1][c] }  (v2f)
//   krow = kk*4 + 2*(lane>>4),  c = nT*16 + (lane&15)
// ---------------------------------------------------------------------------
__global__ void k_packW(const float* __restrict__ W, float* __restrict__ Wf) {
  const int t = (int)(blockIdx.x * blockDim.x + threadIdx.x);   // 0..8191
  if (t >= 32 * 8 * 32) return;
  const int lane = t & 31;
  const int nT   = (t >> 5) & 7;
  const int kk   = t >> 8;
  const int krow = kk * 4 + 2 * (lane >> 4);
  const int c    = nT * 16 + (lane & 15);
  v2f b;
  b.x = W[(size_t)(krow    ) * HxF + c];
  b.y = W[(size_t)(krow + 1) * HxF + c];
  ((v2f*)Wf)[t] = b;
}

// ---------------------------------------------------------------------------
// Kernel 1: h[N,128] = x[N,128] @ W, fp32 WMMA 16x16x4.
// Block = 256 threads = 8 waves; wave owns M-tile (blockIdx.x*8 + wave),
// computes all 8 N-tiles (A fragment reused across 8 WMMAs per K-step).
// ---------------------------------------------------------------------------
__global__ void k_gemm(const float* __restrict__ x, const float* __restrict__ Wf,
                       float* __restrict__ h, int Nn) {
  const int wave   = threadIdx.x >> 5;
  const int lane   = threadIdx.x & 31;
  const int mTile  = (int)blockIdx.x * 8 + wave;
  const int mTiles = (Nn + 15) >> 4;
  if (mTile >= mTiles) return;

  const int mlo   = lane & 15;
  const int khalf = lane >> 4;              // 0 or 1
  int m = mTile * 16 + mlo;
  if (m >= Nn) m = Nn - 1;                  // clamp loads for ragged tail

  // A fragment base: x[m][2*khalf + 4*kk .. +1]  (one b64 per K-step)
  const v2f* __restrict__ xa = (const v2f*)(x + (size_t)m * IN_F + 2 * khalf);
  // Pre-packed B fragments: one b64 per (kk, nT)
  const v2f* __restrict__ wf = ((const v2f*)Wf) + lane;

  v8f acc[8] = {};
  #pragma unroll 2
  for (int kk = 0; kk < 32; ++kk) {         // K = 128, 4 per WMMA
    const v2f a = xa[kk * 2];
    #pragma unroll
    for (int nT = 0; nT < 8; ++nT) {
      const v2f b = wf[(kk * 8 + nT) * 32];
      acc[nT] = __builtin_amdgcn_wmma_f32_16x16x4_f32(
          /*neg_a=*/false, a, /*neg_b=*/false, b,
          /*c_mod=*/(short)0, acc[nT], /*reuse_a=*/false, /*reuse_b=*/false);
    }
  }

  // D layout: VGPR r -> row (r + 8*khalf), col = nT*16 + mlo
  const int rowBase = mTile * 16 + 8 * khalf;
  if (mTile * 16 + 16 <= Nn) {              // wave-uniform fast path
    float* __restrict__ hp = h + (size_t)rowBase * HxF + mlo;
    #pragma unroll
    for (int nT = 0; nT < 8; ++nT)
      #pragma unroll
      for (int r = 0; r < 8; ++r)
        hp[(size_t)r * HxF + nT * 16] = acc[nT][r];
  } else {
    for (int nT = 0; nT < 8; ++nT)
      for (int r = 0; r < 8; ++r) {
        const int row = rowBase + r;
        if (row < Nn) h[(size_t)row * HxF + nT * 16 + mlo] = acc[nT][r];
      }
  }
}

// ---------------------------------------------------------------------------
// Kernel 2: per-node attention coefficients a_src/a_dst [N,4].
// One wave per node: float4 loads, 8-lane xor-shuffle head reductions.
// ---------------------------------------------------------------------------
__global__ void k_attn(const float* __restrict__ h,
                       const float* __restrict__ att_src,
                       const float* __restrict__ att_dst,
                       float* __restrict__ a_src, float* __restrict__ a_dst, int Nn) {
  const int wid  = (int)((blockIdx.x * blockDim.x + threadIdx.x) >> 5);
  const int lane = threadIdx.x & 31;
  if (wid >= Nn) return;

  const float4 hv = ((const float4*)(h + (size_t)wid * HxF))[lane];
  const float4 as = ((const float4*)att_src)[lane];   // [4,32] flat == feature layout
  const float4 ad = ((const float4*)att_dst)[lane];

  float ps = hv.x * as.x + hv.y * as.y + hv.z * as.z + hv.w * as.w;
  float pd = hv.x * ad.x + hv.y * ad.y + hv.z * ad.z + hv.w * ad.w;
  #pragma unroll
  for (int msk = 1; msk < 8; msk <<= 1) {   // reduce within 8-lane head groups
    ps += __shfl_xor(ps, msk);
    pd += __shfl_xor(pd, msk);
  }
  if ((lane & 7) == 0) {
    const int head = lane >> 3;
    a_src[(size_t)wid * HEADS + head] = ps;
    a_dst[(size_t)wid * HEADS + head] = pd;
  }
}

// ---------------------------------------------------------------------------
// Kernel 3: init out=bias, mkey=0 (== -inf in ordered-uint space), s=0.
// ---------------------------------------------------------------------------
__global__ void k_init(float* __restrict__ out, const float* __restrict__ bias,
                       unsigned int* __restrict__ mkey, float* __restrict__ s, int Nn) {
  const int t = (int)(blockIdx.x * blockDim.x + threadIdx.x);
  if (t < Nn * HxF) out[t] = bias[t & (HxF - 1)];
  if (t < Nn * HEADS) { mkey[t] = 0u; s[t] = 0.0f; }
}

// ordered-uint <-> float monotone mapping (exact segment-max via integer atomicMax)
__device__ __forceinline__ unsigned int enc_key(float f) {
  const unsigned int b = __float_as_uint(f);
  return (b & 0x80000000u) ? ~b : (b | 0x80000000u);
}
__device__ __forceinline__ float dec_key(unsigned int k) {
  const unsigned int b = (k & 0x80000000u) ? (k & 0x7FFFFFFFu) : ~k;
  return __uint_as_float(b);
}

// ---------------------------------------------------------------------------
// Kernel 4: edge logits e = leaky_relu(a_src[src]+a_dst[dst]); segment max.
// ---------------------------------------------------------------------------
__global__ void k_edge_logits(const long long* __restrict__ ei,
                              const float* __restrict__ a_src,
                              const float* __restrict__ a_dst,
                              float* __restrict__ ebuf,
                              unsigned int* __restrict__ mkey, int E, int Etot) {
  const int t = (int)(blockIdx.x * blockDim.x + threadIdx.x);
  if (t >= Etot) return;
  int src, dst;
  if (t < E) { src = (int)ei[t]; dst = (int)ei[(size_t)E + t]; }
  else       { src = dst = t - E; }                     // self loop

  const float4 as = ((const float4*)a_src)[src];
  const float4 ad = ((const float4*)a_dst)[dst];
  float4 e;
  e.x = as.x + ad.x; e.y = as.y + ad.y; e.z = as.z + ad.z; e.w = as.w + ad.w;
  e.x = e.x > 0.f ? e.x : NEG_SLOPE * e.x;
  e.y = e.y > 0.f ? e.y : NEG_SLOPE * e.y;
  e.z = e.z > 0.f ? e.z : NEG_SLOPE * e.z;
  e.w = e.w > 0.f ? e.w : NEG_SLOPE * e.w;
  ((float4*)ebuf)[t] = e;

  unsigned int* mk = mkey + (size_t)dst * HEADS;
  atomicMax(&mk[0], enc_key(e.x));
  atomicMax(&mk[1], enc_key(e.y));
  atomicMax(&mk[2], enc_key(e.z));
  atomicMax(&mk[3], enc_key(e.w));
}

// ---------------------------------------------------------------------------
// Kernel 5: ex = exp(e - m[dst]); segment sum s[dst] += ex.  (ex overwrites e)
// ---------------------------------------------------------------------------
__global__ void k_edge_exp(const long long* __restrict__ ei,
                           float* __restrict__ ebuf,
                           const unsigned int* __restrict__ mkey,
                           float* __restrict__ s, int E, int Etot) {
  const int t = (int)(blockIdx.x * blockDim.x + threadIdx.x);
  if (t >= Etot) return;
  int dst;
  if (t < E) dst = (int)ei[(size_t)E + t];
  else       dst = t - E;

  float4 e = ((const float4*)ebuf)[t];
  const uint4 mk = ((const uint4*)mkey)[dst];
  float4 ex;
  ex.x = __expf(e.x - dec_key(mk.x));
  ex.y = __expf(e.y - dec_key(mk.y));
  ex.z = __expf(e.z - dec_key(mk.z));
  ex.w = __expf(e.w - dec_key(mk.w));
  ((float4*)ebuf)[t] = ex;

  float* sp = s + (size_t)dst * HEADS;
  atomicAdd(&sp[0], ex.x);
  atomicAdd(&sp[1], ex.y);
  atomicAdd(&sp[2], ex.z);
  atomicAdd(&sp[3], ex.w);
}

// ---------------------------------------------------------------------------
// Kernel 6: wave per edge: out[dst] += alpha * h[src]  (128 floats / wave,
// float4 per lane; head = lane>>3). fp32 atomics resolve in L2.
// ---------------------------------------------------------------------------
__global__ void k_scatter(const float* __restrict__ h,
                          const long long* __restrict__ ei,
                          const float* __restrict__ ebuf,
                          const float* __restrict__ s,
                          float* __restrict__ out, int E, int Etot) {
  const int wid  = (int)((blockIdx.x * blockDim.x + threadIdx.x) >> 5);
  const int lane = threadIdx.x & 31;
  if (wid >= Etot) return;
  int src, dst;
  if (wid < E) { src = (int)ei[wid]; dst = (int)ei[(size_t)E + wid]; }
  else         { src = dst = wid - E; }

  const int head = lane >> 3;   // lane*4..lane*4+3 all inside one 32-wide head block
  const float alpha = ebuf[(size_t)wid * HEADS + head] /
                      (s[(size_t)dst * HEADS + head] + 1e-16f);

  const float4 hv = ((const float4*)(h + (size_t)src * HxF))[lane];
  float* o = out + (size_t)dst * HxF + lane * 4;
  atomicAdd(o + 0, hv.x * alpha);
  atomicAdd(o + 1, hv.y * alpha);
  atomicAdd(o + 2, hv.z * alpha);
  atomicAdd(o + 3, hv.w * alpha);
}

// ---------------------------------------------------------------------------
extern "C" void kernel_launch(void* const* d_in, const int* in_sizes, int n_in,
                              void* d_out, int out_size, void* d_ws, size_t ws_size,
                              hipStream_t stream) {
  const float*     x       = (const float*)d_in[0];
  const long long* ei      = (const long long*)d_in[1];   // int64 [2,E]
  const float*     W       = (const float*)d_in[2];
  const float*     att_src = (const float*)d_in[3];
  const float*     att_dst = (const float*)d_in[4];
  const float*     bias    = (const float*)d_in[5];
  float*           out     = (float*)d_out;

  const int Nn   = in_sizes[0] / IN_F;   // 100000
  const int E    = in_sizes[1] / 2;      // 1600000
  const int Etot = E + Nn;               // with self loops

  // workspace layout (all 16B-aligned offsets)
  float* ws     = (float*)d_ws;
  float* h      = ws;                                    // Nn*128
  float* a_src  = h     + (size_t)Nn * HxF;              // Nn*4
  float* a_dst  = a_src + (size_t)Nn * HEADS;            // Nn*4
  unsigned int* mkey = (unsigned int*)(a_dst + (size_t)Nn * HEADS); // Nn*4
  float* s      = (float*)(mkey + (size_t)Nn * HEADS);   // Nn*4
  float* ebuf   = s + (size_t)Nn * HEADS;                // Etot*4 (logits, then exp)
  float* Wf     = ebuf + (size_t)Etot * HEADS;           // 16384 (packed W fragments)

  const int mTiles = (Nn + 15) / 16;
  k_packW<<<(8192 + 255) / 256, 256, 0, stream>>>(W, Wf);
  k_gemm<<<(mTiles + 7) / 8, 256, 0, stream>>>(x, Wf, h, Nn);
  k_attn<<<(Nn + 7) / 8, 256, 0, stream>>>(h, att_src, att_dst, a_src, a_dst, Nn);
  k_init<<<((size_t)Nn * HxF + 255) / 256, 256, 0, stream>>>(out, bias, mkey, s, Nn);
  k_edge_logits<<<(Etot + 255) / 256, 256, 0, stream>>>(ei, a_src, a_dst, ebuf, mkey, E, Etot);
  k_edge_exp<<<(Etot + 255) / 256, 256, 0, stream>>>(ei, ebuf, mkey, s, E, Etot);
  k_scatter<<<(Etot + 7) / 8, 256, 0, stream>>>(h, ei, ebuf, s, out, E, Etot);
}